// TripletLoss_57019985822564
// MI455X (gfx1250) — compile-verified
//
#include <hip/hip_runtime.h>
#include <hip/hip_bf16.h>
#include <float.h>

typedef __attribute__((ext_vector_type(2))) float v2f;
typedef __attribute__((ext_vector_type(8))) float v8f;

#define B_N     8192
#define D_DIM   128
#define TILE    16
#define NTILES  (B_N / TILE)      // 512 column tiles
#define CSPLIT  4                 // column chunks per row tile
#define TPC     (NTILES / CSPLIT) // 128 tiles per chunk
#define MARGIN_ 0.3f
#define EPS_    1e-8f

// d_ws layout (4-byte words):
//   [0,      8192)  sq[i]       = ||e_i||^2                (float)
//   [8192,  16384)  hp[i] bits  (running hardest-pos max)  (uint, fp32 bits)
//   [16384, 24576)  hn[i] bits  (running hardest-neg min)  (uint, fp32 bits)
//   [24576]         loss_sum    (float)
//   [24577]         n_triplets  (uint)

__global__ void __launch_bounds__(256)
tl_init_kernel(const float* __restrict__ emb, float* __restrict__ sq,
               unsigned* __restrict__ hp, unsigned* __restrict__ hn,
               float* __restrict__ loss_sum, unsigned* __restrict__ cnt) {
    const int warp = (blockIdx.x * blockDim.x + threadIdx.x) >> 5;
    const int lane = threadIdx.x & 31;
    if (warp < B_N) {
        const float* row = emb + (size_t)warp * D_DIM;
        float4 v = *(const float4*)(row + lane * 4);   // 32 lanes x 4 = 128
        float s = v.x * v.x + v.y * v.y + v.z * v.z + v.w * v.w;
        #pragma unroll
        for (int off = 16; off > 0; off >>= 1) s += __shfl_xor(s, off, 32);
        if (lane == 0) {
            sq[warp] = s;
            hp[warp] = 0u;           // sentinel: no positive found (dist > 0 always)
            hn[warp] = 0x7F7FFFFFu;  // FLT_MAX sentinel: no negative found
        }
    }
    if (blockIdx.x == 0 && threadIdx.x == 0) { *loss_sum = 0.0f; *cnt = 0u; }
}

// One wave per (row-tile, column-chunk). Fused fp32-WMMA Gram tile ->
// distance -> running hardest-pos/neg in registers; cross-wave combine via
// uint-punned float atomics (valid for non-negative floats).
__global__ void __launch_bounds__(256)
tl_gram_hardest_kernel(const float* __restrict__ emb, const int* __restrict__ labels,
                       const float* __restrict__ sq,
                       unsigned* __restrict__ hp_out, unsigned* __restrict__ hn_out) {
    const int wave  = blockIdx.x * (blockDim.x >> 5) + (threadIdx.x >> 5);
    const int lane  = threadIdx.x & 31;
    const int rt    = wave / CSPLIT;
    const int chunk = wave % CSPLIT;
    const int r     = lane & 15;   // row (A) / col (B) within tile
    const int half  = lane >> 4;   // k-pair selector within a 4-wide k step
    const int rowbase = rt * TILE;

    // A fragments for all 32 k-steps of this 16-row tile (reused across cols).
    // V_WMMA_F32_16X16X4_F32 A layout: lane holds row = lane%16,
    // VGPR0/1 = K = 2*(lane/16) + {0,1} within the 4-wide k step.
    v2f a[32];
    const float* arow = emb + (size_t)(rowbase + r) * D_DIM + 2 * half;
    #pragma unroll
    for (int kk = 0; kk < 32; ++kk) a[kk] = *(const v2f*)(arow + 4 * kk);

    // Per-lane accumulator rows: C/D layout -> VGPR v holds row v + 8*half.
    int   lab_r[8]; float sqr[8];
    float hp[8], hn[8];
    #pragma unroll
    for (int v = 0; v < 8; ++v) {
        const int grow = rowbase + v + 8 * half;
        lab_r[v] = labels[grow];
        sqr[v]   = sq[grow];
        hp[v] = 0.0f;
        hn[v] = FLT_MAX;
    }

    const int ct0 = chunk * TPC;
    for (int ct = ct0; ct < ct0 + TPC; ++ct) {
        const int colbase = ct * TILE;
        const int mycol   = colbase + r;   // this lane's column (B + C/D layout)
        const float* brow = emb + (size_t)mycol * D_DIM + 2 * half;

        v8f acc = {};
        #pragma unroll
        for (int kk = 0; kk < 32; ++kk) {
            v2f b = *(const v2f*)(brow + 4 * kk);
            acc = __builtin_amdgcn_wmma_f32_16x16x4_f32(
                      /*neg_a=*/false, a[kk], /*neg_b=*/false, b,
                      /*c_mod=*/(short)0, acc, /*reuse_a=*/false, /*reuse_b=*/false);
        }

        const int   lab_c = labels[mycol];
        const float sqc   = sq[mycol];
        #pragma unroll
        for (int v = 0; v < 8; ++v) {
            const int grow = rowbase + v + 8 * half;
            const float d2 = sqr[v] + sqc - 2.0f * acc[v];
            const float d  = sqrtf(fmaxf(d2, 0.0f) + EPS_);
            if (lab_r[v] == lab_c) {
                if (grow != mycol) hp[v] = fmaxf(hp[v], d);   // positive, not diag
            } else {
                hn[v] = fminf(hn[v], d);                      // negative
            }
        }
    }

    // Columns live across lanes within each 16-lane half; reduce there.
    #pragma unroll
    for (int v = 0; v < 8; ++v) {
        #pragma unroll
        for (int off = 8; off > 0; off >>= 1) {
            hp[v] = fmaxf(hp[v], __shfl_xor(hp[v], off, 32));
            hn[v] = fminf(hn[v], __shfl_xor(hn[v], off, 32));
        }
    }
    if (r == 0) {
        #pragma unroll
        for (int v = 0; v < 8; ++v) {
            const int grow = rowbase + v + 8 * half;
            atomicMax(&hp_out[grow], __float_as_uint(hp[v]));
            atomicMin(&hn_out[grow], __float_as_uint(hn[v]));
        }
    }
}

__global__ void __launch_bounds__(256)
tl_loss_kernel(const unsigned* __restrict__ hp, const unsigned* __restrict__ hn,
               float* __restrict__ loss_sum, unsigned* __restrict__ cnt) {
    const int i = blockIdx.x * blockDim.x + threadIdx.x;
    float per = 0.0f; unsigned c = 0u;
    if (i < B_N) {
        const float p = __uint_as_float(hp[i]);
        const float n = __uint_as_float(hn[i]);
        if (p > 0.0f && n < FLT_MAX) {             // valid anchor
            per = fmaxf(p - n + MARGIN_, 0.0f);
            c = 1u;
        }
    }
    #pragma unroll
    for (int off = 16; off > 0; off >>= 1) {
        per += __shfl_xor(per, off, 32);
        c   += __shfl_xor(c, off, 32);
    }
    if ((threadIdx.x & 31) == 0) {
        atomicAdd(loss_sum, per);
        atomicAdd(cnt, c);
    }
}

__global__ void tl_finalize_kernel(const float* __restrict__ loss_sum,
                                   const unsigned* __restrict__ cnt,
                                   float* __restrict__ out) {
    unsigned c = *cnt;
    if (c < 1u) c = 1u;
    out[0] = *loss_sum / (float)c;
}

extern "C" void kernel_launch(void* const* d_in, const int* in_sizes, int n_in,
                              void* d_out, int out_size, void* d_ws, size_t ws_size,
                              hipStream_t stream) {
    const float* emb    = (const float*)d_in[0];
    const int*   labels = (const int*)d_in[1];

    float*    sq       = (float*)d_ws;
    unsigned* hp       = (unsigned*)d_ws + 8192;
    unsigned* hn       = (unsigned*)d_ws + 16384;
    float*    loss_sum = (float*)d_ws + 24576;
    unsigned* cnt      = (unsigned*)d_ws + 24577;
    float*    out      = (float*)d_out;

    // 1) ||e_i||^2 + sentinel init (d_ws must be re-initialized every call)
    tl_init_kernel<<<B_N / 8, 256, 0, stream>>>(emb, sq, hp, hn, loss_sum, cnt);

    // 2) fused fp32-WMMA Gram + hardest pos/neg: 512 row tiles x 4 col chunks
    const int total_waves = NTILES * CSPLIT;          // 2048 waves
    tl_gram_hardest_kernel<<<total_waves / 8, 256, 0, stream>>>(emb, labels, sq, hp, hn);

    // 3) per-anchor loss + valid count
    tl_loss_kernel<<<B_N / 256, 256, 0, stream>>>(hp, hn, loss_sum, cnt);

    // 4) mean over valid triplets
    tl_finalize_kernel<<<1, 1, 0, stream>>>(loss_sum, cnt, out);
}